// MolecularHamiltonianExactOpt_33380485824796
// MI455X (gfx1250) — compile-verified
//
#include <hip/hip_runtime.h>

#define BB 512
#define NQ 28
#define TT 8192
#define UU 1024

typedef __attribute__((ext_vector_type(2))) float v2f;
typedef __attribute__((ext_vector_type(8))) float v8f;

// ---------------------------------------------------------------------------
// Kernel 1: per-term precompute.
//   c[t]      = cmul(coefficients[t], i^{num_pauli_y[t] & 3})
//   selmask[t]= bitmask of select_idx[t,:]
//   uu[t]     = (u32) unique_indices[t]
// ---------------------------------------------------------------------------
__global__ void __launch_bounds__(256) prep_terms_kernel(
    const long long* __restrict__ uidx64,
    const int* __restrict__ selidx,
    const int* __restrict__ npy,
    const float* __restrict__ coeff,
    float2* __restrict__ cvec,
    unsigned* __restrict__ selmask,
    unsigned* __restrict__ uu) {
  const int t = blockIdx.x * blockDim.x + threadIdx.x;
  if (t >= TT) return;
  unsigned m = 0;
  const int* s = selidx + t * NQ;
#pragma unroll
  for (int q = 0; q < NQ; ++q) m |= ((unsigned)(s[q] & 1)) << q;
  selmask[t] = m;
  uu[t] = (unsigned)uidx64[t];
  const int k = npy[t] & 3;
  const float cr0 = coeff[2 * t], ci0 = coeff[2 * t + 1];
  float cr, ci;
  if (k == 0)      { cr =  cr0; ci =  ci0; }
  else if (k == 1) { cr = -ci0; ci =  cr0; }
  else if (k == 2) { cr = -cr0; ci = -ci0; }
  else             { cr =  ci0; ci = -cr0; }
  cvec[t] = make_float2(cr, ci);
}

// ---------------------------------------------------------------------------
// Kernel 2: d[b,u] = sum_q states[b,q] * (-2*flips[u,q]) * W[q]   (r and i)
// via V_WMMA_F32_16X16X4_F32 (f32 in/out -> exact reference numerics),
// then ratio[b,u] = exp(d_r) * (cos d_i, sin d_i).
// One wave per 16x16 (M=batch, N=u) tile; K=28 = 7 chained K=4 WMMAs x2.
// ---------------------------------------------------------------------------
__global__ void __launch_bounds__(32) wmma_ratio_kernel(
    const int* __restrict__ states,
    const int* __restrict__ flips,
    const float* __restrict__ Wr,
    const float* __restrict__ Wi,
    float2* __restrict__ ratio) {
  const int tile = blockIdx.x;
  const int b0 = (tile >> 6) << 4;  // (tile / (UU/16)) * 16
  const int u0 = (tile & 63) << 4;  // (tile % (UU/16)) * 16
  const int lane = threadIdx.x;     // wave32: one wave per block
  const int half = lane >> 4;       // 0: K pair {k,k+1}, 1: {k+2,k+3}
  const int lm   = lane & 15;       // row (A) / col (B) within tile

  const int* __restrict__ arow = states + (b0 + lm) * NQ;  // A: M = lm
  const int* __restrict__ brow = flips  + (u0 + lm) * NQ;  // B: N = lm

  v8f accR = {};
  v8f accI = {};
#pragma unroll
  for (int k = 0; k < NQ; k += 4) {
    const int kb = k + half * 2;
    v2f A, Br, Bi;
    // A 16x4 layout: lanes0-15 VGPR{0,1}=K{0,1}; lanes16-31 = K{2,3}
    A[0] = (float)arow[kb];
    A[1] = (float)arow[kb + 1];
    // B 4x16 layout mirrors A along K
    const float f0 = -2.0f * (float)brow[kb];
    const float f1 = -2.0f * (float)brow[kb + 1];
    Br[0] = f0 * Wr[kb];
    Br[1] = f1 * Wr[kb + 1];
    Bi[0] = f0 * Wi[kb];
    Bi[1] = f1 * Wi[kb + 1];
    accR = __builtin_amdgcn_wmma_f32_16x16x4_f32(
        false, A, false, Br, (short)0, accR, false, false);
    accI = __builtin_amdgcn_wmma_f32_16x16x4_f32(
        false, A, false, Bi, (short)0, accI, false, false);
  }

  // C/D layout: VGPR r holds row M=r (lanes0-15) / M=r+8 (lanes16-31), N=lane&15
#pragma unroll
  for (int r = 0; r < 8; ++r) {
    const int m = r + half * 8;
    const float dr = accR[r];
    const float di = accI[r];
    const float mag = expf(dr);
    float sn, cs;
    sincosf(di, &sn, &cs);
    ratio[(size_t)(b0 + m) * UU + (u0 + lm)] = make_float2(mag * cs, mag * sn);
  }
}

// ---------------------------------------------------------------------------
// Kernel 3: per-batch reduction over T terms.
//   sign(b,t) = (-1)^popcount(xneg_mask[b] & selmask[t])   (x = -states)
//   out[b]   += sign * cmul(c[t], ratio[b, uu[t]])
// ratio row (1024 x float2 = 8KB) staged in LDS, one block per b.
// ---------------------------------------------------------------------------
__global__ void __launch_bounds__(256) reduce_kernel(
    const int* __restrict__ states,
    const float2* __restrict__ ratio,
    const float2* __restrict__ cvec,
    const unsigned* __restrict__ selmask,
    const unsigned* __restrict__ uu,
    float* __restrict__ out) {
  __shared__ float2 lr[UU];
  __shared__ float sred[256];
  __shared__ float sired[256];
  const int b = blockIdx.x;
  const int tid = threadIdx.x;

  for (int i = tid; i < UU; i += 256) lr[i] = ratio[(size_t)b * UU + i];

  unsigned xneg = 0;  // bits where x = -states is negative, i.e. states > 0
  const int* __restrict__ srow = states + b * NQ;
#pragma unroll
  for (int q = 0; q < NQ; ++q) xneg |= ((unsigned)(srow[q] > 0)) << q;
  __syncthreads();

  float ar = 0.0f, ai = 0.0f;
  for (int t = tid; t < TT; t += 256) {
    const float2 c = cvec[t];
    const float2 r = lr[uu[t]];
    const float s = (__popc(xneg & selmask[t]) & 1) ? -1.0f : 1.0f;
    ar += s * (c.x * r.x - c.y * r.y);
    ai += s * (c.x * r.y + c.y * r.x);
  }
  sred[tid] = ar;
  sired[tid] = ai;
  __syncthreads();
  for (int off = 128; off > 0; off >>= 1) {
    if (tid < off) {
      sred[tid] += sred[tid + off];
      sired[tid] += sired[tid + off];
    }
    __syncthreads();
  }
  if (tid == 0) {
    out[2 * b]     = sred[0];
    out[2 * b + 1] = sired[0];
  }
}

// ---------------------------------------------------------------------------
extern "C" void kernel_launch(void* const* d_in, const int* in_sizes, int n_in,
                              void* d_out, int out_size, void* d_ws, size_t ws_size,
                              hipStream_t stream) {
  const int*       states = (const int*)d_in[0];        // (B,NQ) int32 {-1,+1}
  const int*       flips  = (const int*)d_in[1];        // (U,NQ) int32 {0,1}
  const long long* uidx64 = (const long long*)d_in[2];  // (T,) int64
  const int*       selidx = (const int*)d_in[3];        // (T,NQ) int32 {0,1}
  const int*       npy    = (const int*)d_in[4];        // (T,) int32
  const float*     coeff  = (const float*)d_in[5];      // (T,2) f32
  const float*     Wr     = (const float*)d_in[6];      // (NQ,) f32
  const float*     Wi     = (const float*)d_in[7];      // (NQ,) f32
  float*           out    = (float*)d_out;              // (B,2) f32

  char* ws = (char*)d_ws;
  float2* ratio = (float2*)ws;                             // B*U*8   = 4 MB
  size_t off = (size_t)BB * UU * sizeof(float2);
  float2* cvec = (float2*)(ws + off);  off += (size_t)TT * sizeof(float2);   // 64 KB
  unsigned* selmask = (unsigned*)(ws + off); off += (size_t)TT * sizeof(unsigned); // 32 KB
  unsigned* uu = (unsigned*)(ws + off);                    // 32 KB

  prep_terms_kernel<<<TT / 256, 256, 0, stream>>>(uidx64, selidx, npy, coeff,
                                                  cvec, selmask, uu);
  wmma_ratio_kernel<<<(BB / 16) * (UU / 16), 32, 0, stream>>>(states, flips,
                                                              Wr, Wi, ratio);
  reduce_kernel<<<BB, 256, 0, stream>>>(states, ratio, cvec, selmask, uu, out);
}